// GNN_83923660964216
// MI455X (gfx1250) — compile-verified
//
#include <hip/hip_runtime.h>
#include <hip/hip_bf16.h>
#include <math.h>

// ---------------------------------------------------------------------------
// GCN forward for MI455X (gfx1250, wave32).
// GEMMs: V_WMMA_F32_16X16X4_F32 (exact fp32 matrix path), 64x128 block tile,
// A/B chunks staged into LDS with global_load_async_to_lds_b128 (ASYNCcnt).
// Edge aggregation: global_atomic_add_f32; the 51MB zw table is L2-resident.
// ---------------------------------------------------------------------------

#define GN_N 50000
#define GN_E 800000
#define GN_D 256
#define GN_L 4

typedef float v2f __attribute__((ext_vector_type(2)));
typedef float v8f __attribute__((ext_vector_type(8)));

// ---------------------------------------------------------------------------
// degree / normalization coefficient kernels
// ---------------------------------------------------------------------------
__global__ __launch_bounds__(256) void gnn_deg_init(float* __restrict__ deg) {
    int i = blockIdx.x * 256 + threadIdx.x;
    if (i < GN_N) deg[i] = 1.0f;  // self-loop weight
}

__global__ __launch_bounds__(256) void gnn_deg_edges(const int* __restrict__ dst,
                                                     const float* __restrict__ w,
                                                     float* __restrict__ deg) {
    int e = blockIdx.x * 256 + threadIdx.x;
    if (e < GN_E) atomicAdd(&deg[dst[e]], w[e]);
}

__global__ __launch_bounds__(256) void gnn_dinv(float* __restrict__ deg) {
    int i = blockIdx.x * 256 + threadIdx.x;
    if (i >= GN_N) return;
    float d = deg[i];
    deg[i] = (d > 0.0f) ? rsqrtf(fmaxf(d, 1e-12f)) : 0.0f;  // in-place -> dinv
}

__global__ __launch_bounds__(256) void gnn_coef(const int* __restrict__ src,
                                                const int* __restrict__ dst,
                                                const float* __restrict__ w,
                                                const float* __restrict__ dinv,
                                                float* __restrict__ coef) {
    int e = blockIdx.x * 256 + threadIdx.x;
    if (e < GN_E) coef[e] = dinv[src[e]] * w[e] * dinv[dst[e]];
}

__global__ __launch_bounds__(256) void gnn_coef_self(const float* __restrict__ dinv,
                                                     float* __restrict__ cs) {
    int i = blockIdx.x * 256 + threadIdx.x;
    if (i < GN_N) cs[i] = dinv[i] * dinv[i];  // self-loop: weight 1
}

// ---------------------------------------------------------------------------
// LayerNorm: one wave (32 lanes) per 256-float row; 8 rows per block.
// ---------------------------------------------------------------------------
__global__ __launch_bounds__(256) void gnn_layernorm(const float* __restrict__ H,
                                                     const float* __restrict__ lw,
                                                     const float* __restrict__ lb,
                                                     float* __restrict__ Z) {
    const int lane = threadIdx.x & 31;
    const int row  = blockIdx.x * 8 + (threadIdx.x >> 5);
    if (row >= GN_N) return;
    const float* hp = H + (size_t)row * GN_D;

    float4 v0 = *(const float4*)(hp + lane * 4);
    float4 v1 = *(const float4*)(hp + 128 + lane * 4);

    float s = v0.x + v0.y + v0.z + v0.w + v1.x + v1.y + v1.z + v1.w;
#pragma unroll
    for (int o = 16; o > 0; o >>= 1) s += __shfl_xor(s, o, 32);
    const float mu = s * (1.0f / 256.0f);

    float dx, ss = 0.0f;
    dx = v0.x - mu; ss += dx * dx;  dx = v0.y - mu; ss += dx * dx;
    dx = v0.z - mu; ss += dx * dx;  dx = v0.w - mu; ss += dx * dx;
    dx = v1.x - mu; ss += dx * dx;  dx = v1.y - mu; ss += dx * dx;
    dx = v1.z - mu; ss += dx * dx;  dx = v1.w - mu; ss += dx * dx;
#pragma unroll
    for (int o = 16; o > 0; o >>= 1) ss += __shfl_xor(ss, o, 32);
    const float rs = rsqrtf(ss * (1.0f / 256.0f) + 1e-5f);

    float4 w0 = *(const float4*)(lw + lane * 4);
    float4 w1 = *(const float4*)(lw + 128 + lane * 4);
    float4 b0 = *(const float4*)(lb + lane * 4);
    float4 b1 = *(const float4*)(lb + 128 + lane * 4);

    float4 o0, o1;
    o0.x = (v0.x - mu) * rs * w0.x + b0.x;  o0.y = (v0.y - mu) * rs * w0.y + b0.y;
    o0.z = (v0.z - mu) * rs * w0.z + b0.z;  o0.w = (v0.w - mu) * rs * w0.w + b0.w;
    o1.x = (v1.x - mu) * rs * w1.x + b1.x;  o1.y = (v1.y - mu) * rs * w1.y + b1.y;
    o1.z = (v1.z - mu) * rs * w1.z + b1.z;  o1.w = (v1.w - mu) * rs * w1.w + b1.w;

    float* zp = Z + (size_t)row * GN_D;
    *(float4*)(zp + lane * 4) = o0;
    *(float4*)(zp + 128 + lane * 4) = o1;
}

// ---------------------------------------------------------------------------
// fp32 WMMA GEMM: C[M x 256] = A[M x 256] @ B[256 x 256] (+ bias)
// 64x128 tile per 256-thread block; K staged through LDS in 64-wide chunks
// via global_load_async_to_lds_b128. 8 waves; wave w owns rowTile (w>>1) and
// colTiles {4*(w&1) .. 4*(w&1)+3} (4 accumulators -> 4x A-fragment reuse).
// ---------------------------------------------------------------------------
#define TM 64
#define TNB 128
#define TK 64
#define LDA 72   // padded LDS stride for As (floats); 72*4 B is 16B-aligned
#define LDB 132  // padded LDS stride for Bs (floats); 132*4 B is 16B-aligned

__device__ __forceinline__ void gnn_async_b128(void* lds, const float* g) {
    const unsigned l = (unsigned)(size_t)lds;           // LDS byte offset
    const unsigned long long a = (unsigned long long)(size_t)g;
    asm volatile("global_load_async_to_lds_b128 %0, %1, off"
                 :: "v"(l), "v"(a) : "memory");
}

__global__ __launch_bounds__(256) void gnn_gemm_wmma(const float* __restrict__ A,
                                                     const float* __restrict__ B,
                                                     const float* __restrict__ bias,
                                                     float* __restrict__ C,
                                                     int M) {
    __shared__ float As[TM][LDA];
    __shared__ float Bs[TK][LDB];

    const int tid  = threadIdx.x;
    const int lane = tid & 31;
    const int wv   = tid >> 5;           // 0..7
    const int rowT = wv >> 1;            // 0..3
    const int colG = (wv & 1) * 4;       // 0 or 4 (first of 4 col tiles)
    const int rowBase = blockIdx.x * TM;
    const int colBase = blockIdx.y * TNB;

    v8f acc[4] = {};

    for (int k0 = 0; k0 < GN_D; k0 += TK) {
        // ---- stage A chunk (64 x 64) : async global -> LDS, 16B per lane ----
#pragma unroll
        for (int i = 0; i < 4; ++i) {
            const int idx = tid + 256 * i;       // 0..1023 float4 slots
            const int r   = idx >> 4;            // 0..63
            const int c4  = (idx & 15) << 2;     // 0..60 step 4
            const int gr  = rowBase + r;
            if (gr < M) {
                gnn_async_b128(&As[r][c4], A + (size_t)gr * GN_D + k0 + c4);
            } else {
                *(float4*)(&As[r][c4]) = make_float4(0.f, 0.f, 0.f, 0.f);
            }
        }
        // ---- stage B chunk (64 x 128) : async global -> LDS ----
#pragma unroll
        for (int i = 0; i < 8; ++i) {
            const int idx = tid + 256 * i;       // 0..2047 float4 slots
            const int r   = idx >> 5;            // 0..63
            const int c4  = (idx & 31) << 2;     // 0..124 step 4
            gnn_async_b128(&Bs[r][c4], B + (size_t)(k0 + r) * GN_D + colBase + c4);
        }
        asm volatile("s_wait_asynccnt 0" ::: "memory");
        __syncthreads();

        const int m    = lane & 15;
        const int koff = (lane >> 4) << 1;  // 0 (lanes 0-15) or 2 (lanes 16-31)
#pragma unroll
        for (int kk = 0; kk < TK; kk += 4) {
            // A fragment (16x4 f32): v0 = A[m][k+koff], v1 = A[m][k+koff+1]
            v2f a;
            a.x = As[rowT * 16 + m][kk + koff];
            a.y = As[rowT * 16 + m][kk + koff + 1];
#pragma unroll
            for (int ct = 0; ct < 4; ++ct) {
                v2f b;
                b.x = Bs[kk + koff][(colG + ct) * 16 + m];
                b.y = Bs[kk + koff + 1][(colG + ct) * 16 + m];
                acc[ct] = __builtin_amdgcn_wmma_f32_16x16x4_f32(
                    false, a, false, b, (short)0, acc[ct], false, false);
            }
        }
        __syncthreads();
    }

    // C/D layout: element r of v8f holds M = r + 8*(lane>>4), N = lane&15
    const int n  = lane & 15;
    const int mo = (lane >> 4) << 3;
#pragma unroll
    for (int ct = 0; ct < 4; ++ct) {
        const int col  = colBase + (colG + ct) * 16 + n;
        const float bv = bias ? bias[col] : 0.0f;
#pragma unroll
        for (int r = 0; r < 8; ++r) {
            const int grow = rowBase + rowT * 16 + r + mo;
            if (grow < M) C[(size_t)grow * GN_D + col] = acc[ct][r] + bv;
        }
    }
}

// ---------------------------------------------------------------------------
// Aggregation: agg initialized with self-loop term, then edge scatter-add.
// ---------------------------------------------------------------------------
__global__ __launch_bounds__(256) void gnn_agg_init(const float* __restrict__ zw,
                                                    const float* __restrict__ cs,
                                                    float* __restrict__ agg) {
    int t = blockIdx.x * 256 + threadIdx.x;      // one float4 per thread
    if (t >= GN_N * (GN_D / 4)) return;
    const int row = t >> 6;
    const int c4  = (t & 63) << 2;
    const float c = cs[row];
    float4 v = *(const float4*)(zw + (size_t)row * GN_D + c4);
    float4 o = make_float4(c * v.x, c * v.y, c * v.z, c * v.w);
    *(float4*)(agg + (size_t)row * GN_D + c4) = o;
}

__global__ __launch_bounds__(256) void gnn_edge_agg(const float* __restrict__ zw,
                                                    float* __restrict__ agg,
                                                    const int* __restrict__ src,
                                                    const int* __restrict__ dst,
                                                    const float* __restrict__ coef) {
    // 64 lanes per edge, float4 per lane -> 256 floats per edge
    const long long t = (long long)blockIdx.x * 256 + threadIdx.x;
    const int e = (int)(t >> 6);
    if (e >= GN_E) return;
    const int g  = (int)(t & 63);
    const int s  = src[e];
    const int d0 = dst[e];
    const float c = coef[e];
    float4 v = *(const float4*)(zw + (size_t)s * GN_D + g * 4);
    float* base = agg + (size_t)d0 * GN_D + g * 4;
    atomicAdd(base + 0, c * v.x);
    atomicAdd(base + 1, c * v.y);
    atomicAdd(base + 2, c * v.z);
    atomicAdd(base + 3, c * v.w);
}

// ---------------------------------------------------------------------------
// bias + exact GELU + residual
// ---------------------------------------------------------------------------
__global__ __launch_bounds__(256) void gnn_bias_gelu_res(const float* __restrict__ agg,
                                                         const float* __restrict__ bias,
                                                         const float* __restrict__ hPrev,
                                                         float* __restrict__ hOut) {
    int i = blockIdx.x * 256 + threadIdx.x;
    if (i >= GN_N * GN_D) return;
    const int col = i & (GN_D - 1);
    const float x = agg[i] + bias[col];
    const float g = 0.5f * x * (1.0f + erff(x * 0.70710678118654752440f));
    hOut[i] = g + hPrev[i];
}

// ---------------------------------------------------------------------------
// host launcher
// ---------------------------------------------------------------------------
extern "C" void kernel_launch(void* const* d_in, const int* in_sizes, int n_in,
                              void* d_out, int out_size, void* d_ws, size_t ws_size,
                              hipStream_t stream) {
    (void)in_sizes; (void)n_in; (void)out_size; (void)ws_size;

    const float* x_unused  = (const float*)d_in[0]; (void)x_unused;
    const int*   eidx      = (const int*)d_in[1];         // [2, E]
    const float* ew        = (const float*)d_in[2];       // [E]
    const float* emb       = (const float*)d_in[3];       // [N, D]
    const float* ln_w      = (const float*)d_in[4];       // [L, D]
    const float* ln_b      = (const float*)d_in[5];       // [L, D]
    const float* conv_w    = (const float*)d_in[6];       // [L, D, D]
    const float* conv_b    = (const float*)d_in[7];       // [L, D]
    const float* post_w    = (const float*)d_in[8];       // [D, D]
    const float* post_b    = (const float*)d_in[9];       // [D]

    const int* src = eidx;          // edge_index[0]
    const int* dst = eidx + GN_E;   // edge_index[1]

    // workspace layout (floats)
    float* ws    = (float*)d_ws;
    float* dinv  = ws;                               // N  (deg -> dinv in place)
    float* coef  = dinv + GN_N;                      // E
    float* cself = coef + GN_E;                      // N
    float* h     = cself + GN_N;                     // N*D
    float* z     = h + (size_t)GN_N * GN_D;          // N*D  (aliased as agg)
    float* zw    = z + (size_t)GN_N * GN_D;          // N*D
    float* agg   = z;  // safe alias: GEMM (reads z) completes before agg_init

    const int blkN  = (GN_N + 255) / 256;
    const int blkE  = (GN_E + 255) / 256;
    const int blkLN = (GN_N + 7) / 8;
    const int blkEA = (GN_E * 64 + 255) / 256;
    const int blkND = (GN_N * GN_D + 255) / 256;
    const dim3 gemmGrid((GN_N + TM - 1) / TM, GN_D / TNB);

    // graph normalization coefficients
    gnn_deg_init<<<blkN, 256, 0, stream>>>(dinv);
    gnn_deg_edges<<<blkE, 256, 0, stream>>>(dst, ew, dinv);
    gnn_dinv<<<blkN, 256, 0, stream>>>(dinv);
    gnn_coef<<<blkE, 256, 0, stream>>>(src, dst, ew, dinv, coef);
    gnn_coef_self<<<blkN, 256, 0, stream>>>(dinv, cself);

    for (int l = 0; l < GN_L; ++l) {
        const float* hin = (l == 0) ? emb : h;
        gnn_layernorm<<<blkLN, 256, 0, stream>>>(hin, ln_w + l * GN_D,
                                                 ln_b + l * GN_D, z);
        gnn_gemm_wmma<<<gemmGrid, 256, 0, stream>>>(
            z, conv_w + (size_t)l * GN_D * GN_D, nullptr, zw, GN_N);
        gnn_agg_init<<<(GN_N * 64 + 255) / 256, 256, 0, stream>>>(zw, cself, agg);
        gnn_edge_agg<<<blkEA, 256, 0, stream>>>(zw, agg, src, dst, coef);
        gnn_bias_gelu_res<<<blkND, 256, 0, stream>>>(agg, conv_b + l * GN_D, hin, h);
    }

    // final projection with fused bias
    gnn_gemm_wmma<<<gemmGrid, 256, 0, stream>>>(h, post_w, post_b,
                                                (float*)d_out, GN_N);
}